// SymmetricChannel_30468497998502
// MI455X (gfx1250) — compile-verified
//
#include <hip/hip_runtime.h>

// ---------------------------------------------------------------------------
// SymmetricChannel (Gumbel-Softmax noisy channel), MI455X / gfx1250.
//
// Shapes: B=4096, L=20, V=256  ->  N = 81920 rows of 256 floats.
// Memory-bound: ~504 MB of traffic -> ~21.6 us floor @ 23.3 TB/s.
// Strategy: 1 row per 256-thread block (8 wave32s).
//   - async-copy the messages/probs row into LDS (ASYNCcnt) while each lane
//     computes its hash RNG (mask + replacement column) in VALU,
//   - intra-row scatter of moved mass via ds_add_f32 LDS atomics,
//   - elementwise probs update (needs p[row,0] broadcast -> read from LDS),
//   - pure copies streamed back out with async LDS->global stores.
// ---------------------------------------------------------------------------

#define VCOLS 256
// P(mask) = 0.1 as a 32-bit uint threshold: 0.1 * 2^32
#define MASK_THRESH 429496730u

// The gfx1250 async-copy builtins take addrspace-qualified int* operands
// (clang prints AS1 as "__device__ int *" in HIP mode).
#define AS1_INT(p) ((__attribute__((address_space(1))) int*)(p))
#define AS3_INT(p) ((__attribute__((address_space(3))) int*)(p))

__device__ __forceinline__ unsigned int mix32(unsigned int x) {
  // low-bias 32-bit finalizer (deterministic per-element RNG)
  x ^= x >> 16; x *= 0x7feb352du;
  x ^= x >> 15; x *= 0x846ca68bu;
  x ^= x >> 16;
  return x;
}

__global__ __launch_bounds__(VCOLS) void symmetric_channel_kernel(
    const float* __restrict__ gmsg,   // [N, 256]
    const float* __restrict__ gprob,  // [N, 256]
    float* __restrict__ out_mnoisy,   // [N, 256]
    float* __restrict__ out_mcopy,    // [N, 256]
    float* __restrict__ out_pnoisy,   // [N, 256]
    float* __restrict__ out_pcopy,    // [N, 256]
    float* __restrict__ out_eos)      // [N]
{
  __shared__ float s_m[VCOLS];
  __shared__ float s_p[VCOLS];
  __shared__ float s_adj[VCOLS];

  const int tid = threadIdx.x;                 // column index 0..255
  const long long row = blockIdx.x;            // row index 0..N-1
  const long long idx = row * VCOLS + tid;

  // ---- Stage 1: async global->LDS copy of this row (ASYNCcnt path) --------
#if __has_builtin(__builtin_amdgcn_global_load_async_to_lds_b32)
  __builtin_amdgcn_global_load_async_to_lds_b32(
      AS1_INT(gmsg + idx), AS3_INT(s_m + tid), /*offset=*/0, /*cpol=*/0);
  __builtin_amdgcn_global_load_async_to_lds_b32(
      AS1_INT(gprob + idx), AS3_INT(s_p + tid), /*offset=*/0, /*cpol=*/0);
#else
  s_m[tid] = gmsg[idx];
  s_p[tid] = gprob[idx];
#endif

  // ---- Stage 2: per-lane RNG, overlapped with the async copy --------------
  // Column tid (tid>=1) is flipped with prob 0.1; its mass goes to a uniform
  // random non-EOS column != tid.
  unsigned moved_mask = 0u;
  unsigned repl = (unsigned)tid;
  if (tid >= 1) {
    const unsigned e  = (unsigned)(row * (VCOLS - 1) + (tid - 1));
    const unsigned h1 = mix32(e ^ 0x2545F491u);  // Bernoulli stream
    const unsigned h2 = mix32(e + 0x9E3779B9u);  // replacement-id stream
    moved_mask = (h1 < MASK_THRESH) ? 1u : 0u;
    // rid uniform in [0, 254); map to replacement column in 1..255, != tid
    const unsigned rid =
        (unsigned)(((unsigned long long)h2 * (VCOLS - 2ull)) >> 32);
    repl = (rid + 1u < (unsigned)tid) ? (rid + 1u) : (rid + 2u);
  }

  // ---- Wait for our async loads, then make LDS visible block-wide ---------
#if __has_builtin(__builtin_amdgcn_s_wait_asynccnt)
  __builtin_amdgcn_s_wait_asynccnt(0);
#elif defined(__gfx1250__)
  asm volatile("s_wait_asynccnt 0" ::: "memory");
#endif
  __syncthreads();

  const float m   = s_m[tid];
  const float pr  = s_p[tid];
  const float pr0 = s_p[0];                    // per-row EOS prob broadcast

  // ---- Stage 3: intra-row scatter of moved mass via LDS -------------------
  // adj[tid] starts at -moved (unique owner write), then masked lanes
  // scatter +moved into adj[repl] with ds_add_f32 atomics (~10% lanes live).
  s_adj[tid] = (tid >= 1 && moved_mask) ? -m : 0.0f;
  __syncthreads();
  if (tid >= 1 && moved_mask) {
    atomicAdd(&s_adj[repl], m);                // -> ds_add_f32
  }
  __syncthreads();

  // ---- Stage 4: results ---------------------------------------------------
  out_mnoisy[idx] = m + s_adj[tid];

  // probs_noisy: j==0 passthrough; j>=1: 0.9*p + (1-p-p0)*(0.1/254)
  const float padj = fmaf(pr, 0.9f, (1.0f - pr - pr0) * (0.1f / 254.0f));
  out_pnoisy[idx] = (tid == 0) ? pr : padj;

  // Pure copies: stream straight from LDS to memory (no VGPR round-trip).
#if __has_builtin(__builtin_amdgcn_global_store_async_from_lds_b32)
  __builtin_amdgcn_global_store_async_from_lds_b32(
      AS1_INT(out_mcopy + idx), AS3_INT(s_m + tid), /*offset=*/0, /*cpol=*/0);
  __builtin_amdgcn_global_store_async_from_lds_b32(
      AS1_INT(out_pcopy + idx), AS3_INT(s_p + tid), /*offset=*/0, /*cpol=*/0);
  __builtin_amdgcn_s_wait_asynccnt(0);
#else
  out_mcopy[idx] = m;
  out_pcopy[idx] = pr;
#endif

  if (tid == 0) out_eos[row] = 0.0f;           // accumulated_eos_prob = zeros
}

extern "C" void kernel_launch(void* const* d_in, const int* in_sizes, int n_in,
                              void* d_out, int out_size, void* d_ws, size_t ws_size,
                              hipStream_t stream) {
  const float* gmsg  = (const float*)d_in[0];  // messages, B*L*V floats
  const float* gprob = (const float*)d_in[1];  // probs,    B*L*V floats
  float* out = (float*)d_out;

  const long long S = (long long)in_sizes[0];  // B*L*V = 20,971,520
  const long long nrows = S / VCOLS;           // B*L   = 81,920

  float* out_mnoisy = out;
  float* out_mcopy  = out + S;
  float* out_pnoisy = out + 2 * S;
  float* out_pcopy  = out + 3 * S;
  float* out_eos    = out + 4 * S;

  symmetric_channel_kernel<<<dim3((unsigned)nrows), dim3(VCOLS), 0, stream>>>(
      gmsg, gprob, out_mnoisy, out_mcopy, out_pnoisy, out_pcopy, out_eos);
}